// ComplexGaborFilter_51067161150099
// MI455X (gfx1250) — compile-verified
//
#include <hip/hip_runtime.h>
#include <math.h>

#define N_FILTERS 64
#define KLEN      402          // actual taps: arange(-201, 201)
#define KPAD      404          // padded to multiple of 4 for WMMA K-steps
#define L_IN      16000
#define L_OUT     (L_IN - KLEN + 1)   // 15599
#define BATCH     16
#define WAVES_PER_BLOCK 8
#define N_PER_WAVE  32                         // two 16-wide N tiles per wave
#define N_PER_BLOCK (WAVES_PER_BLOCK * N_PER_WAVE)   // 256 output positions
#define XWIN      (N_PER_BLOCK + KPAD)         // 660 floats of x staged per block

typedef __attribute__((ext_vector_type(2))) float v2f;
typedef __attribute__((ext_vector_type(8))) float v8f;

// ---------------------------------------------------------------------------
// Kernel 1: materialize the Gabor filter bank (real & imag), padded to KPAD.
//   kre/kim layout: [N_FILTERS][KPAD], zeros in the pad region.
// ---------------------------------------------------------------------------
__global__ void gabor_filters_kernel(const float* __restrict__ cf,
                                     const float* __restrict__ bwv,
                                     float* __restrict__ kre,
                                     float* __restrict__ kim) {
    int idx = blockIdx.x * blockDim.x + threadIdx.x;
    if (idx >= N_FILTERS * KPAD) return;
    int f = idx / KPAD;
    int k = idx - f * KPAD;
    float re = 0.0f, im = 0.0f;
    if (k < KLEN) {
        float t  = (float)(k - 201);            // t in [-201, 200]
        float b  = bwv[f];
        float env = __expf(-t * t / (2.0f * b * b)) * (1.0f / (2.5066282746310002f * b));
        float ph  = cf[f] * t;
        re = env * __cosf(ph);
        im = env * __sinf(ph);
    }
    kre[idx] = re;
    kim[idx] = im;
}

// ---------------------------------------------------------------------------
// Kernel 2: implicit-GEMM Gabor conv via V_WMMA_F32_16X16X4_F32.
//   Each wave owns a 16(M filters) x 32(N positions) complex tile:
//   4 accumulators (re/im x 2 N-tiles). Per K-step: 2 LDS b64 A-loads,
//   4 LDS b32 B-loads, 4 WMMAs — A fragments reused across both N tiles.
// ---------------------------------------------------------------------------
__global__ void __launch_bounds__(256, 2)
gabor_conv_wmma(const float* __restrict__ x,
                const float* __restrict__ kre,
                const float* __restrict__ kim,
                float* __restrict__ out) {
    __shared__ float s_kre[16 * KPAD];
    __shared__ float s_kim[16 * KPAD];
    __shared__ float s_x[XWIN + 4];

    const int tid   = threadIdx.x;
    const int lane  = tid & 31;
    const int wave  = tid >> 5;
    const int nb    = blockIdx.x * N_PER_BLOCK;   // block's base output position
    const int ftile = blockIdx.y * 16;            // filter tile base
    const int bidx  = blockIdx.z;                 // batch index

    const float* xb = x + (size_t)bidx * L_IN;

    // Stage 16-filter tile (re+im) into LDS. Tile rows are contiguous.
    const int tile_base = ftile * KPAD;
    for (int i = tid; i < 16 * KPAD; i += 256) {
        s_kre[i] = kre[tile_base + i];
        s_kim[i] = kim[tile_base + i];
    }
    // Stage x window [nb, nb + XWIN) with zero fill past the signal end.
    for (int i = tid; i < XWIN; i += 256) {
        int gi = nb + i;
        s_x[i] = (gi < L_IN) ? xb[gi] : 0.0f;
    }
    __syncthreads();

    // Fragment addressing (ISA 16x16x4 f32 layouts):
    //   A: VGPR0 = K {0 | 2}, VGPR1 = K {1 | 3} for lanes {0-15 | 16-31},
    //      row M = lane & 15. B mirrors with column N = lane & 15.
    const int half  = lane >> 4;          // 0 or 1
    const int koff  = half << 1;          // 0 or 2
    const int m     = lane & 15;
    const int abase = (m * KPAD + koff) >> 1;          // v2f index (8B aligned)
    const int bbase = wave * N_PER_WAVE + (lane & 15) + koff;

    // v2f views of the filter LDS arrays: guaranteed 8-byte-aligned b64 loads.
    const v2f* s_kre2 = (const v2f*)s_kre;
    const v2f* s_kim2 = (const v2f*)s_kim;

    v8f c_re0 = {}, c_re1 = {};
    v8f c_im0 = {}, c_im1 = {};

#pragma unroll 4
    for (int kk = 0; kk < KPAD; kk += 4) {
        v2f a_re = s_kre2[abase + (kk >> 1)];
        v2f a_im = s_kim2[abase + (kk >> 1)];
        v2f bv0, bv1;
        bv0.x = s_x[bbase + kk];
        bv0.y = s_x[bbase + kk + 1];
        bv1.x = s_x[bbase + kk + 16];
        bv1.y = s_x[bbase + kk + 17];
        c_re0 = __builtin_amdgcn_wmma_f32_16x16x4_f32(
                    false, a_re, false, bv0, (short)0, c_re0, false, false);
        c_re1 = __builtin_amdgcn_wmma_f32_16x16x4_f32(
                    false, a_re, false, bv1, (short)0, c_re1, false, false);
        c_im0 = __builtin_amdgcn_wmma_f32_16x16x4_f32(
                    false, a_im, false, bv0, (short)0, c_im0, false, false);
        c_im1 = __builtin_amdgcn_wmma_f32_16x16x4_f32(
                    false, a_im, false, bv1, (short)0, c_im1, false, false);
    }

    // C layout: VGPR v -> row (v + 8*half), column = lane & 15.
    const int mbase = half << 3;
    const int npos0 = nb + wave * N_PER_WAVE + (lane & 15);
    const int npos1 = npos0 + 16;
    float2* outc = (float2*)out;   // interleaved complex64
    size_t rowbase = ((size_t)(bidx * N_FILTERS + ftile + mbase)) * L_OUT;
    if (npos0 < L_OUT) {
#pragma unroll
        for (int v = 0; v < 8; ++v)
            outc[rowbase + (size_t)v * L_OUT + npos0] = make_float2(c_re0[v], c_im0[v]);
    }
    if (npos1 < L_OUT) {
#pragma unroll
        for (int v = 0; v < 8; ++v)
            outc[rowbase + (size_t)v * L_OUT + npos1] = make_float2(c_re1[v], c_im1[v]);
    }
}

// ---------------------------------------------------------------------------
extern "C" void kernel_launch(void* const* d_in, const int* in_sizes, int n_in,
                              void* d_out, int out_size, void* d_ws, size_t ws_size,
                              hipStream_t stream) {
    const float* x  = (const float*)d_in[0];   // [16, 1, 16000] f32
    const float* cf = (const float*)d_in[1];   // [64] f32
    const float* bw = (const float*)d_in[2];   // [64] f32
    float* out = (float*)d_out;                // [16, 64, 15599] complex64 (interleaved)

    float* kre = (float*)d_ws;                 // [64][KPAD] f32
    float* kim = kre + N_FILTERS * KPAD;       // [64][KPAD] f32

    // 1) Build padded filter bank in workspace.
    int total = N_FILTERS * KPAD;
    gabor_filters_kernel<<<(total + 255) / 256, 256, 0, stream>>>(cf, bw, kre, kim);

    // 2) WMMA implicit-GEMM convolution.
    dim3 grid((L_OUT + N_PER_BLOCK - 1) / N_PER_BLOCK, N_FILTERS / 16, BATCH);
    gabor_conv_wmma<<<grid, 256, 0, stream>>>(x, kre, kim, out);
}